// GCN_6408091206361
// MI455X (gfx1250) — compile-verified
//
#include <hip/hip_runtime.h>
#include <hip/hip_bf16.h>

typedef __attribute__((ext_vector_type(2))) float v2f;
typedef __attribute__((ext_vector_type(8))) float v8f;

static constexpr int FD = 64;   // feature dim

// ---------------------------------------------------------------------------
// Utility kernels
// ---------------------------------------------------------------------------
__global__ void k_fill(float* p, float v, int n) {
    int i = blockIdx.x * blockDim.x + threadIdx.x;
    if (i < n) p[i] = v;
}

// deg[c] += ew[e] (self-loop "1" already in init value)
__global__ void k_deg_acc(const int* __restrict__ col,
                          const float* __restrict__ ew,
                          float* __restrict__ deg, int E) {
    int e = blockIdx.x * blockDim.x + threadIdx.x;
    if (e < E) atomicAdd(&deg[col[e]], ew[e]);
}

// dis = deg > 0 ? rsqrt(deg) : 0   (in place)
__global__ void k_rsqrt(float* __restrict__ deg, int n) {
    int i = blockIdx.x * blockDim.x + threadIdx.x;
    if (i < n) {
        float d = deg[i];
        deg[i] = (d > 0.0f) ? rsqrtf(d) : 0.0f;
    }
}

// ---------------------------------------------------------------------------
// WMMA fp32 GEMM: Y[n_rows x 64] = X[n_rows x 64] @ W[64 x 64]
// One wave per 16-row tile. n_rows must be a multiple of 16 (50000 = 3125*16).
// A-layout (16x4 f32): lane L<16 holds row M=L, v0=K0,v1=K1; lanes 16-31 K2,K3.
// B-layout (4x16 f32): lane L holds col N=L&15; lane half selects K-pair.
// C/D-layout: VGPR r, lanes 0-15 -> M=r, N=lane; lanes 16-31 -> M=r+8.
// ---------------------------------------------------------------------------
__global__ void k_gemm64_wmma(const float* __restrict__ X,
                              const float* __restrict__ W,
                              float* __restrict__ Y, int n_rows) {
    int wave = (int)((blockIdx.x * blockDim.x + threadIdx.x) >> 5);
    int lane = threadIdx.x & 31;
    int m0 = wave * 16;
    if (m0 >= n_rows) return;               // wave-uniform: EXEC stays all-1s

    const int mlo = lane & 15;
    const int hi2 = (lane >> 4) << 1;       // 0 for lanes 0-15, 2 for 16-31

    // Load the full A tile for this 16-row block: 16 K-steps of float2.
    v2f a[16];
    const float* xrow = X + (size_t)(m0 + mlo) * FD;
    #pragma unroll
    for (int kk = 0; kk < 16; ++kk) {
        int c0 = 4 * kk + hi2;
        a[kk].x = xrow[c0];
        a[kk].y = xrow[c0 + 1];
    }

    #pragma unroll
    for (int nt = 0; nt < 4; ++nt) {
        const int ncol = nt * 16 + mlo;
        v8f c = {};
        #pragma unroll
        for (int kk = 0; kk < 16; ++kk) {
            int k0 = 4 * kk + hi2;
            v2f b;
            b.x = W[(size_t)k0 * FD + ncol];
            b.y = W[(size_t)(k0 + 1) * FD + ncol];
            c = __builtin_amdgcn_wmma_f32_16x16x4_f32(
                    /*neg_a=*/false, a[kk], /*neg_b=*/false, b,
                    /*c_mod=*/(short)0, c, /*reuse_a=*/false, /*reuse_b=*/false);
        }
        float* yb = Y + (size_t)(m0 + ((lane >> 4) << 3)) * FD + ncol;
        #pragma unroll
        for (int r = 0; r < 8; ++r)
            yb[(size_t)r * FD] = c[r];
    }
}

// ---------------------------------------------------------------------------
// Edge scatter: acc[c,:] += xw[r,:] * (dis[r]*ew*dis[c]).  64 threads / edge.
// ---------------------------------------------------------------------------
__global__ void k_scatter(const int* __restrict__ row,
                          const int* __restrict__ col,
                          const float* __restrict__ ew,
                          const float* __restrict__ dis,
                          const float* __restrict__ xw,
                          float* __restrict__ acc, int E) {
    long long t = (long long)blockIdx.x * blockDim.x + threadIdx.x;
    int e = (int)(t >> 6);
    if (e >= E) return;
    int d = (int)(t & 63);
    int r = row[e], c = col[e];
    float nrm = dis[r] * ew[e] * dis[c];
    atomicAdd(&acc[(size_t)c * FD + d], xw[(size_t)r * FD + d] * nrm);
}

// h = relu(acc + xw*dis^2 + b)   (self-loop term fused here)
__global__ void k_finish(const float* __restrict__ acc,
                         const float* __restrict__ xw,
                         const float* __restrict__ dis,
                         const float* __restrict__ b,
                         float* __restrict__ h, int n_nodes) {
    long long t = (long long)blockIdx.x * blockDim.x + threadIdx.x;
    int i = (int)(t >> 6);
    if (i >= n_nodes) return;
    int d = (int)(t & 63);
    size_t idx = (size_t)i * FD + d;
    float s = dis[i];
    float v = acc[idx] + xw[idx] * s * s + b[d];
    h[idx] = fmaxf(v, 0.0f);
}

// ---------------------------------------------------------------------------
// Mean pooling over graph ids
// ---------------------------------------------------------------------------
__global__ void k_pool_acc(const float* __restrict__ h,
                           const int* __restrict__ batch,
                           float* __restrict__ sums,
                           float* __restrict__ cnt, int n_nodes) {
    long long t = (long long)blockIdx.x * blockDim.x + threadIdx.x;
    int i = (int)(t >> 6);
    if (i >= n_nodes) return;
    int d = (int)(t & 63);
    int g = batch[i];
    atomicAdd(&sums[(size_t)g * FD + d], h[(size_t)i * FD + d]);
    if (d == 0) atomicAdd(&cnt[g], 1.0f);
}

__global__ void k_pool_div(const float* __restrict__ sums,
                           const float* __restrict__ cnt,
                           float* __restrict__ out, int total) {
    int t = blockIdx.x * blockDim.x + threadIdx.x;
    if (t >= total) return;
    int g = t >> 6;
    out[t] = sums[t] / fmaxf(cnt[g], 1.0f);
}

// ---------------------------------------------------------------------------
// Launcher
// ---------------------------------------------------------------------------
extern "C" void kernel_launch(void* const* d_in, const int* in_sizes, int n_in,
                              void* d_out, int out_size, void* d_ws, size_t ws_size,
                              hipStream_t stream) {
    const float* x   = (const float*)d_in[0];      // [N, 64]
    const int*   ei  = (const int*)  d_in[1];      // [2, E]
    const float* ew  = (const float*)d_in[2];      // [E]
    const int*   bat = (const int*)  d_in[3];      // [N]
    const float* W1  = (const float*)d_in[4];
    const float* b1  = (const float*)d_in[5];
    const float* W2  = (const float*)d_in[6];
    const float* b2  = (const float*)d_in[7];

    const int N = in_sizes[3];          // 50000
    const int E = in_sizes[2];          // 800000
    const int G = out_size / FD;        // 64
    const int* row = ei;
    const int* col = ei + E;

    // Workspace carve-up (floats)
    float* ws   = (float*)d_ws;
    float* dis  = ws;                               // N   (deg -> rsqrt in place)
    float* xw   = dis + N;                          // N*64
    float* acc  = xw  + (size_t)N * FD;             // N*64
    float* h    = acc + (size_t)N * FD;             // N*64
    float* sums = h   + (size_t)N * FD;             // G*64
    float* cnt  = sums + (size_t)G * FD;            // G
    (void)ws_size; (void)n_in;

    const int B = 256;
    const int gN    = (N + B - 1) / B;
    const int gE    = (E + B - 1) / B;
    const int gNF   = (int)(((long long)N * FD + B - 1) / B);
    const int gEF   = (int)(((long long)E * FD + B - 1) / B);
    const int waves = (N + 15) / 16;                 // one wave per 16-row tile
    const int gGemm = (waves + (B / 32) - 1) / (B / 32);

    // --- degree / normalization (shared by both layers) ---
    k_fill<<<gN, B, 0, stream>>>(dis, 1.0f, N);                 // self-loop weight
    k_deg_acc<<<gE, B, 0, stream>>>(col, ew, dis, E);
    k_rsqrt<<<gN, B, 0, stream>>>(dis, N);

    // --- layer 1 ---
    k_gemm64_wmma<<<gGemm, B, 0, stream>>>(x, W1, xw, N);
    k_fill<<<gNF, B, 0, stream>>>(acc, 0.0f, N * FD);
    k_scatter<<<gEF, B, 0, stream>>>(row, col, ew, dis, xw, acc, E);
    k_finish<<<gNF, B, 0, stream>>>(acc, xw, dis, b1, h, N);

    // --- layer 2 ---
    k_gemm64_wmma<<<gGemm, B, 0, stream>>>(h, W2, xw, N);
    k_fill<<<gNF, B, 0, stream>>>(acc, 0.0f, N * FD);
    k_scatter<<<gEF, B, 0, stream>>>(row, col, ew, dis, xw, acc, E);
    k_finish<<<gNF, B, 0, stream>>>(acc, xw, dis, b2, h, N);    // in-place over h: ok

    // --- mean pooling ---
    k_fill<<<(G * FD + G + B - 1) / B, B, 0, stream>>>(sums, 0.0f, G * FD + G);
    k_pool_acc<<<gNF, B, 0, stream>>>(h, bat, sums, cnt, N);
    k_pool_div<<<(G * FD + B - 1) / B, B, 0, stream>>>(sums, cnt, (float*)d_out, G * FD);
}